// NSVFPointSampler_2327872274948
// MI455X (gfx1250) — compile-verified
//
#include <hip/hip_runtime.h>

// ---- problem constants (from reference) ----
#define N_RAYS    32768
#define MAX_HITS  64
#define MAX_STEPS 704
#define STEP_SIZE 0.005f
#define DET_NOISE 0.5f
#define WPB       8               // waves per block (wave32 => 256 threads)
#define CHUNKS    (MAX_STEPS / 32) // 22 chunks of 32 steps per ray

typedef float v2f __attribute__((ext_vector_type(2)));
typedef float v8f __attribute__((ext_vector_type(8)));

__device__ __forceinline__ float wave_sum32(float x) {
#pragma unroll
  for (int d = 16; d >= 1; d >>= 1) x += __shfl_xor(x, d, 32);
  return x;
}

__device__ __forceinline__ float wave_iscan32(float x, int lane) {
#pragma unroll
  for (int d = 1; d < 32; d <<= 1) {
    float y = __shfl_up(x, d, 32);
    if (lane >= d) x += y;
  }
  return x;
}

// t(j) for one ray: u = min((j+0.5)/steps, 1-1e-6); searchsorted-right into cdf;
// tvals = t_near[b] + (u - cdf0[b]) / (p > 0 ? p : 1) * seg[b].  Also returns bin b.
__device__ __forceinline__ float sample_t(int j, float stepsf,
                                          const float* __restrict__ cdfext, // 65 entries, cdfext[0]=0
                                          const float* __restrict__ probs,
                                          const float* __restrict__ tn,
                                          const float* __restrict__ sg,
                                          int* __restrict__ bout) {
  float u = ((float)j + DET_NOISE) / stepsf;
  u = fminf(u, 1.0f - 1e-6f);
  // count of cdf entries <= u  (cdf[i] == cdfext[i+1]), 6 LDS probes
  int b = 0;
#pragma unroll
  for (int s = 32; s >= 1; s >>= 1)
    b += (cdfext[b + s] <= u) ? s : 0;
  b = (b > MAX_HITS - 1) ? (MAX_HITS - 1) : b;
  float clo = cdfext[b];
  float p   = probs[b];
  float pp  = (p > 0.0f) ? p : 1.0f;
  *bout = b;
  return tn[b] + (u - clo) / pp * sg[b];
}

__global__ __launch_bounds__(256) void nsvf_cdf_sample_kernel(
    const float* __restrict__ rays_o, const float* __restrict__ rays_d,
    const int* __restrict__ vox_idx, const float* __restrict__ t_near,
    const float* __restrict__ t_far,
    float* __restrict__ out_pts, int* __restrict__ out_svidx,
    float* __restrict__ out_tvals, float* __restrict__ out_dists) {
  __shared__ float s_cdfext[WPB][MAX_HITS + 2]; // [0]=0.0, [1..64]=cdf
  __shared__ float s_probs [WPB][MAX_HITS];
  __shared__ float s_tnear [WPB][MAX_HITS];
  __shared__ float s_seg   [WPB][MAX_HITS];     // raw t_far - t_near (unmasked)
  __shared__ int   s_vox   [WPB][MAX_HITS];

  const int lane = threadIdx.x & 31;
  const int w    = threadIdx.x >> 5;
  const int ray  = blockIdx.x * WPB + w;

  // ---- stage this ray's 64 hits: 2 per lane ----
  const size_t rb = (size_t)ray * MAX_HITS;
  float tn_a = t_near[rb + lane];        float tn_b = t_near[rb + lane + 32];
  float tf_a = t_far [rb + lane];        float tf_b = t_far [rb + lane + 32];
  int   vx_a = vox_idx[rb + lane];       int   vx_b = vox_idx[rb + lane + 32];

  float sr_a = tf_a - tn_a,              sr_b = tf_b - tn_b;
  float sm_a = (vx_a >= 0) ? sr_a : 0.f, sm_b = (vx_b >= 0) ? sr_b : 0.f;

  float t_range = wave_sum32(sm_a + sm_b);
  float pr_a = sm_a / t_range,           pr_b = sm_b / t_range;
  float sc_a = wave_iscan32(pr_a, lane);
  float totA = __shfl(sc_a, 31, 32);
  float sc_b = wave_iscan32(pr_b, lane) + totA;

  int   steps  = (int)ceilf(t_range / STEP_SIZE);
  float stepsf = (float)steps;

  if (lane == 0) s_cdfext[w][0] = 0.0f;
  s_cdfext[w][lane + 1]  = sc_a;   s_cdfext[w][lane + 33] = sc_b;
  s_probs [w][lane]      = pr_a;   s_probs [w][lane + 32] = pr_b;
  s_tnear [w][lane]      = tn_a;   s_tnear [w][lane + 32] = tn_b;
  s_seg   [w][lane]      = sr_a;   s_seg   [w][lane + 32] = sr_b;
  s_vox   [w][lane]      = vx_a;   s_vox   [w][lane + 32] = vx_b;
  // all LDS traffic is intra-wave; DS ops are in-order per wave -> no barrier

  const float* cdfext = &s_cdfext[w][0];
  const float* probs  = &s_probs [w][0];
  const float* tnear  = &s_tnear [w][0];
  const float* seg    = &s_seg   [w][0];

  // ---- WMMA operands invariant over the ray ----
  // B (4x16 f32): row K=0 carries rays_d over cols 0..2; all other K rows zero.
  // C (16x16 f32): every row = rays_o broadcast (cols 0..2), so D = t*d + o.
  const int col = lane & 15;
  float dval = (lane < 3) ? rays_d[(size_t)ray * 3 + lane] : 0.0f; // lanes 0..2 only
  float oval = (col  < 3) ? rays_o[(size_t)ray * 3 + col ] : 0.0f; // both lane halves
  v2f bmat; bmat[0] = (lane < 16) ? dval : 0.0f; bmat[1] = 0.0f;
  v8f cmat;
#pragma unroll
  for (int k = 0; k < 8; ++k) cmat[k] = oval;

  // ---- 22 chunks of 32 steps ----
  for (int c = 0; c < CHUNKS; ++c) {
    const int j0 = c * 32;
    const int j  = j0 + lane;

    int b, bp, bn;
    float t   = sample_t(j, stepsf, cdfext, probs, tnear, seg, &b);
    float tp  = sample_t((j > 0) ? j - 1 : 0, stepsf, cdfext, probs, tnear, seg, &bp);
    float tnx = sample_t(j + 1, stepsf, cdfext, probs, tnear, seg, &bn);
    tp  = (j > 0)         ? tp  : t;   // t_prev clamp at front
    tnx = (j + 1 < steps) ? tnx : t;   // is_last clamp at back

    const bool  inr = (j < steps);
    const float tv  = inr ? t : 0.0f;
    const float dv  = inr ? 0.5f * (tnx - tp) : 0.0f;
    const int   sv  = inr ? s_vox[w][b] : -1;

    const size_t oidx = (size_t)ray * MAX_STEPS + j;
    out_tvals[oidx] = tv;
    out_dists[oidx] = dv;
    out_svidx[oidx] = sv;

    // ---- pts via V_WMMA_F32_16X16X4_F32:  D = A(t) x B(d) + C(o) ----
    // A col K=0 = masked tvals; WMMA0 rows j0..j0+15 (lanes 0..15 own t),
    // WMMA1 rows j0+16..j0+31 (shuffle upper half's t into lanes 0..15).
    float thi = __shfl(tv, (lane & 15) + 16, 32);
    v2f a0; a0[0] = (lane < 16) ? tv  : 0.0f; a0[1] = 0.0f;
    v2f a1; a1[0] = (lane < 16) ? thi : 0.0f; a1[1] = 0.0f;

    v8f d0 = __builtin_amdgcn_wmma_f32_16x16x4_f32(
        false, a0, false, bmat, (short)0, cmat, false, false);
    v8f d1 = __builtin_amdgcn_wmma_f32_16x16x4_f32(
        false, a1, false, bmat, (short)0, cmat, false, false);

    // D layout: VGPR k -> row k (lanes 0..15) / row k+8 (lanes 16..31), col = lane&15
    if (col < 3) {
      const int rowbase = (lane < 16) ? 0 : 8;
      const size_t pbase = ((size_t)ray * MAX_STEPS + j0) * 3 + col;
#pragma unroll
      for (int k = 0; k < 8; ++k) {
        out_pts[pbase + (size_t)(rowbase + k) * 3]      = d0[k];
        out_pts[pbase + (size_t)(rowbase + k + 16) * 3] = d1[k];
      }
    }
  }
}

extern "C" void kernel_launch(void* const* d_in, const int* in_sizes, int n_in,
                              void* d_out, int out_size, void* d_ws, size_t ws_size,
                              hipStream_t stream) {
  const float* rays_o  = (const float*)d_in[0];
  const float* rays_d  = (const float*)d_in[1];
  const int*   vox_idx = (const int*)  d_in[2];
  const float* t_near  = (const float*)d_in[3];
  const float* t_far   = (const float*)d_in[4];

  float* out = (float*)d_out;
  const size_t P3 = (size_t)N_RAYS * MAX_STEPS * 3; // pts
  const size_t S  = (size_t)N_RAYS * MAX_STEPS;     // per-sample arrays
  float* out_pts   = out;
  int*   out_svidx = (int*)(out + P3);
  float* out_tvals = out + P3 + S;
  float* out_dists = out + P3 + 2 * S;

  dim3 grid(N_RAYS / WPB), block(32 * WPB);
  nsvf_cdf_sample_kernel<<<grid, block, 0, stream>>>(
      rays_o, rays_d, vox_idx, t_near, t_far,
      out_pts, out_svidx, out_tvals, out_dists);
}